// MiniGPT_1194000908433
// MI455X (gfx1250) — compile-verified
//
#include <hip/hip_runtime.h>
#include <hip/hip_bf16.h>

// ---------------- model constants (match reference) ----------------
#define VOCAB 50257
#define EMB   2048
#define NHEAD 16
#define HDIM  128           // EMB / NHEAD
#define TCTX  8192          // cached KV rows
#define VROWS 64            // V rows per TDM tile (64*128*4B = 32KB LDS)
#define ATT_SCALE 0.08838834764831845f   // 1/sqrt(128)

typedef __attribute__((ext_vector_type(2))) float v2f;
typedef __attribute__((ext_vector_type(8))) float v8f;
typedef __attribute__((ext_vector_type(4))) unsigned int v4u;
typedef __attribute__((ext_vector_type(8))) unsigned int v8u;

// =====================================================================
// Embedding + double RMSNorm:  x1 = rms(wte[tok]+wpe[pos]);  xn = rms(x1)
// Single block, 256 threads, LDS tree reductions.
// =====================================================================
__global__ void k_embed_rms(const int* __restrict__ tok, const int* __restrict__ pos,
                            const float* __restrict__ wte, const float* __restrict__ wpe,
                            float* __restrict__ x_res1, float* __restrict__ x_norm) {
    __shared__ float red[256];
    __shared__ float s_inv;
    const int tid = threadIdx.x;
    const int t = *tok, p = *pos;
    const float* a = wte + (size_t)t * EMB;
    const float* b = wpe + (size_t)p * EMB;

    float ss = 0.f;
    for (int i = tid; i < EMB; i += 256) { float v = a[i] + b[i]; ss += v * v; }
    red[tid] = ss; __syncthreads();
    for (int s = 128; s; s >>= 1) { if (tid < s) red[tid] += red[tid + s]; __syncthreads(); }
    if (tid == 0) s_inv = rsqrtf(red[0] / (float)EMB + 1e-5f);
    __syncthreads();
    const float inv1 = s_inv;
    __syncthreads();

    ss = 0.f;
    for (int i = tid; i < EMB; i += 256) {
        float v = (a[i] + b[i]) * inv1;
        x_res1[i] = v;
        ss += v * v;
    }
    red[tid] = ss; __syncthreads();
    for (int s = 128; s; s >>= 1) { if (tid < s) red[tid] += red[tid + s]; __syncthreads(); }
    if (tid == 0) s_inv = rsqrtf(red[0] / (float)EMB + 1e-5f);
    __syncthreads();
    const float inv2 = s_inv;
    for (int i = tid; i < EMB; i += 256) x_norm[i] = x_res1[i] * inv2;
}

// =====================================================================
// Generic RMSNorm: out = in * rsqrt(mean(in^2)+eps). One block, 256 thr.
// =====================================================================
__global__ void k_rms(const float* __restrict__ in, float* __restrict__ out, int n) {
    __shared__ float red[256];
    __shared__ float s_inv;
    const int tid = threadIdx.x;
    float ss = 0.f;
    for (int i = tid; i < n; i += 256) { float v = in[i]; ss += v * v; }
    red[tid] = ss; __syncthreads();
    for (int s = 128; s; s >>= 1) { if (tid < s) red[tid] += red[tid + s]; __syncthreads(); }
    if (tid == 0) s_inv = rsqrtf(red[0] / (float)n + 1e-5f);
    __syncthreads();
    const float inv = s_inv;
    for (int i = tid; i < n; i += 256) out[i] = in[i] * inv;
}

// =====================================================================
// WMMA f32 GEMV: y[r] = sum_c W[r,c] * x[c]  (+res, optional relu)
// One block per 16-row tile; 8 waves split K; V_WMMA_F32_16X16X4_F32
// with B = x broadcast across N. Inner loop unrolled 4x (16 K / body),
// plus global_prefetch_b8 ~4KB ahead on the weight stream.
// n_cols must be a multiple of 128. Rows clamped so EXEC stays all-1s.
// =====================================================================
__global__ void k_gemv_wmma(const float* __restrict__ W, const float* __restrict__ x,
                            const float* __restrict__ res, float* __restrict__ y,
                            int n_rows, int n_cols, int do_relu) {
    __shared__ float part[8 * 16];
    const int tile = blockIdx.x;
    const int lane = threadIdx.x & 31;
    const int wave = threadIdx.x >> 5;
    const int half = lane >> 4;          // 0: K lo pair, 1: K hi pair
    const int m    = lane & 15;          // row within tile

    int row = tile * 16 + m;
    if (row >= n_rows) row = n_rows - 1; // clamp loads; store is predicated
    const float* wrow = W + (size_t)row * (size_t)n_cols;

    const int kw = n_cols >> 3;          // K per wave (8 waves); mult of 16
    const int k0 = wave * kw;

    v8f c;
#pragma unroll
    for (int i = 0; i < 8; ++i) c[i] = 0.f;

    for (int k = k0; k < k0 + kw; k += 16) {
        __builtin_prefetch(wrow + k + 1024, 0, 0);
#pragma unroll
        for (int u = 0; u < 16; u += 4) {
            const int kk = k + u + 2 * half;
            v2f a = *(const v2f*)(wrow + kk);   // A: 16x4 weight tile
            v2f b = *(const v2f*)(x + kk);      // B: x chunk, broadcast over N
            c = __builtin_amdgcn_wmma_f32_16x16x4_f32(false, a, false, b,
                                                      (short)0, c, false, false);
        }
    }

    if (lane == 0 || lane == 16) {
        const int base = wave * 16 + half * 8;
#pragma unroll
        for (int i = 0; i < 8; ++i) part[base + i] = c[i];
    }
    __syncthreads();

    if (threadIdx.x < 16) {
        float s = 0.f;
#pragma unroll
        for (int w = 0; w < 8; ++w) s += part[w * 16 + threadIdx.x];
        const int r = tile * 16 + (int)threadIdx.x;
        if (r < n_rows) {
            if (res) s += res[r];
            if (do_relu) s = fmaxf(s, 0.f);
            y[r] = s;
        }
    }
}

// =====================================================================
// Attention scores: att[t,h] = SCALE * dot(K[t, h*128:(h+1)*128], q_h)
// One wave per (t,h); lanes hold float4 chunks; wave shuffle reduce.
// =====================================================================
__global__ void k_att_scores(const float* __restrict__ keys, const float* __restrict__ kvec,
                             const float* __restrict__ q, float* __restrict__ att) {
    const int gw   = (int)((blockIdx.x * blockDim.x + threadIdx.x) >> 5);
    const int lane = threadIdx.x & 31;
    const int total = (TCTX + 1) * NHEAD;
    if (gw >= total) return;                 // uniform per wave
    const int t = gw / NHEAD, h = gw % NHEAD;
    const float* krow = (t < TCTX) ? (keys + (size_t)t * EMB + h * HDIM)
                                   : (kvec + h * HDIM);
    const float* qh = q + h * HDIM;
    const float4 kv = *(const float4*)(krow + lane * 4);
    const float4 qv = *(const float4*)(qh + lane * 4);
    float s = kv.x * qv.x + kv.y * qv.y + kv.z * qv.z + kv.w * qv.w;
    for (int off = 16; off; off >>= 1) s += __shfl_down(s, off, 32);
    if (lane == 0) att[(size_t)t * NHEAD + h] = s * ATT_SCALE;
}

// =====================================================================
// Per-head softmax over time (stable), in place; also zeroes x_attn[h].
// One block per head.
// =====================================================================
__global__ void k_softmax(float* __restrict__ att, float* __restrict__ x_attn) {
    __shared__ float red[256];
    __shared__ float s_val;
    const int h = blockIdx.x, tid = threadIdx.x;
    const int n = TCTX + 1;

    float m = -1e30f;
    for (int t = tid; t < n; t += 256) m = fmaxf(m, att[(size_t)t * NHEAD + h]);
    red[tid] = m; __syncthreads();
    for (int s = 128; s; s >>= 1) { if (tid < s) red[tid] = fmaxf(red[tid], red[tid + s]); __syncthreads(); }
    if (tid == 0) s_val = red[0];
    __syncthreads();
    m = s_val;
    __syncthreads();

    float sum = 0.f;
    for (int t = tid; t < n; t += 256) {
        const size_t idx = (size_t)t * NHEAD + h;
        float e = __expf(att[idx] - m);
        att[idx] = e;
        sum += e;
    }
    red[tid] = sum; __syncthreads();
    for (int s = 128; s; s >>= 1) { if (tid < s) red[tid] += red[tid + s]; __syncthreads(); }
    if (tid == 0) s_val = red[0];
    __syncthreads();
    const float inv = 1.0f / s_val;
    for (int t = tid; t < n; t += 256) att[(size_t)t * NHEAD + h] *= inv;

    if (tid < HDIM) x_attn[h * HDIM + tid] = 0.f;   // prep for atomic accum
}

// =====================================================================
// Weighted V accumulation via Tensor Data Mover:
//   x_attn[h*128+d] += sum_t w[t,h] * V[t, h*128+d]
// Each block owns (head h, time chunk). Wave 0 programs a TDM descriptor
// (D#) for a VROWSx128-f32 2D tile of V (row stride EMB) and issues
// tensor_load_to_lds; TENSORcnt tracks completion (s_wait_tensorcnt 0),
// then all 4 waves accumulate the tile from LDS. Current-step v row
// (t == TCTX) handled in the scalar tail. tchunk % VROWS == 0.
// =====================================================================
__global__ void k_att_v(const float* __restrict__ values, const float* __restrict__ vvec,
                        const float* __restrict__ att, float* __restrict__ x_attn,
                        int tchunk) {
    __shared__ float vtile[VROWS * HDIM];           // 32 KB
    const int h = blockIdx.y;
    const int d = threadIdx.x;                      // 0..127
    const int n = TCTX + 1;
    int t0 = blockIdx.x * tchunk;
    int t1 = t0 + tchunk; if (t1 > n) t1 = n;
    const int t1c = (t1 > TCTX) ? TCTX : t1;        // cached-region end
    float acc = 0.f;

    // low 32 bits of the generic shared-memory address == LDS byte offset
    const unsigned int lds_base = (unsigned int)(unsigned long long)(void*)vtile;

    for (int tb = t0; tb < t1c; tb += VROWS) {
        int rows = t1c - tb; if (rows > VROWS) rows = VROWS;

        if (threadIdx.x < 32) {                     // wave 0 drives the DMA
            const unsigned long long gaddr =
                (unsigned long long)(const void*)(values + (size_t)tb * EMB + h * HDIM);

            // ---- D# group 0 (count / lds_addr / global_addr / type=2) ----
            v4u g0;
            g0[0] = 1u;                                         // count=1
            g0[1] = lds_base;                                   // lds_addr
            g0[2] = (unsigned int)gaddr;                        // global_addr[31:0]
            g0[3] = (unsigned int)((gaddr >> 32) & 0x1FFFFFFull)// global_addr[56:32]
                  | (2u << 30);                                 // type=2 ("image")

            // ---- D# group 1 ----
            v8u g1;
            g1[0] = (2u << 16);                 // wg_mask=0, data_size=2 (4B), no pad/iter
            g1[1] = ((unsigned int)EMB & 0xFFFFu) << 16;        // tensor_dim0[15:0]=2048
            g1[2] = (((unsigned int)EMB >> 16) & 0xFFFFu)       // tensor_dim0[31:16]=0
                  | (((unsigned int)rows & 0xFFFFu) << 16);     // tensor_dim1[15:0]=rows
            g1[3] = (((unsigned int)rows >> 16) & 0xFFFFu)      // tensor_dim1[31:16]=0
                  | ((unsigned int)HDIM << 16);                 // tile_dim0=128
            g1[4] = (unsigned int)rows & 0xFFFFu;               // tile_dim1=rows, tile_dim2=0
            g1[5] = (unsigned int)EMB;          // tensor_dim0_stride[31:0]=2048
            g1[6] = 0u;                         // dim0_stride[47:32]=0, dim1_stride lo=0
            g1[7] = 0u;                         // tensor_dim1_stride hi = 0

            asm volatile("tensor_load_to_lds %0, %1"
                         :: "s"(g0), "s"(g1) : "memory");
            asm volatile("s_wait_tensorcnt 0x0" ::: "memory");
        }
        __syncthreads();                            // tile visible to all waves

        for (int r = 0; r < rows; ++r)
            acc += att[(size_t)(tb + r) * NHEAD + h] * vtile[r * HDIM + d];
        __syncthreads();                            // protect tile before next DMA
    }

    // tail: current-step v row (t == TCTX), if inside this chunk
    for (int t = t1c; t < t1; ++t)
        acc += att[(size_t)t * NHEAD + h] * vvec[h * HDIM + d];

    atomicAdd(&x_attn[h * HDIM + d], acc);
}

// =====================================================================
// Host-side orchestration
// =====================================================================
extern "C" void kernel_launch(void* const* d_in, const int* in_sizes, int n_in,
                              void* d_out, int out_size, void* d_ws, size_t ws_size,
                              hipStream_t stream) {
    const int*   tok    = (const int*)  d_in[0];
    const int*   pos    = (const int*)  d_in[1];
    const float* keys   = (const float*)d_in[2];
    const float* values = (const float*)d_in[3];
    const float* wte    = (const float*)d_in[4];
    const float* wpe    = (const float*)d_in[5];
    const float* wq     = (const float*)d_in[6];
    const float* wk     = (const float*)d_in[7];
    const float* wv     = (const float*)d_in[8];
    const float* wo     = (const float*)d_in[9];
    const float* w1     = (const float*)d_in[10];
    const float* w2     = (const float*)d_in[11];
    const float* lm_w   = (const float*)d_in[12];
    float* logits = (float*)d_out;

    float* ws = (float*)d_ws;
    float* x_res1 = ws;                 size_t off = EMB;
    float* x_norm = ws + off;           off += EMB;
    float* qv     = ws + off;           off += EMB;
    float* kv     = ws + off;           off += EMB;
    float* vv     = ws + off;           off += EMB;
    float* att    = ws + off;           off += (size_t)(TCTX + 1) * NHEAD;
    float* x_attn = ws + off;           off += EMB;
    float* x2     = ws + off;           off += EMB;
    float* x_nrm2 = ws + off;           off += EMB;
    float* hbuf   = ws + off;           off += 4 * EMB;
    float* x3     = ws + off;           off += EMB;
    (void)ws_size; (void)n_in; (void)in_sizes; (void)out_size;

    // 1) embedding + double RMSNorm
    k_embed_rms<<<1, 256, 0, stream>>>(tok, pos, wte, wpe, x_res1, x_norm);

    // 2) q/k/v GEMVs (WMMA): 2048/16 = 128 tiles each
    k_gemv_wmma<<<128, 256, 0, stream>>>(wq, x_norm, nullptr, qv, EMB, EMB, 0);
    k_gemv_wmma<<<128, 256, 0, stream>>>(wk, x_norm, nullptr, kv, EMB, EMB, 0);
    k_gemv_wmma<<<128, 256, 0, stream>>>(wv, x_norm, nullptr, vv, EMB, EMB, 0);

    // 3) attention scores over 8193 KV rows x 16 heads, one wave each
    {
        const int total_waves = (TCTX + 1) * NHEAD;      // 131088
        const int blocks = (total_waves + 7) / 8;        // 8 waves/block
        k_att_scores<<<blocks, 256, 0, stream>>>(keys, kv, qv, att);
    }

    // 4) per-head softmax (+ zero x_attn)
    k_softmax<<<NHEAD, 256, 0, stream>>>(att, x_attn);

    // 5) weighted V accumulation (TDM-staged tiles)
    {
        const int tchunk = 512;                          // multiple of VROWS
        const int nchunk = (TCTX + 1 + tchunk - 1) / tchunk;  // 17
        k_att_v<<<dim3(nchunk, NHEAD), 128, 0, stream>>>(values, vv, att, x_attn, tchunk);
    }

    // 6) output projection + residual
    k_gemv_wmma<<<128, 256, 0, stream>>>(wo, x_attn, x_res1, x2, EMB, EMB, 0);

    // 7) RMSNorm
    k_rms<<<1, 256, 0, stream>>>(x2, x_nrm2, EMB);

    // 8) MLP: relu(w1 @ x) then w2 @ h + residual
    k_gemv_wmma<<<512, 256, 0, stream>>>(w1, x_nrm2, nullptr, hbuf, 4 * EMB, EMB, 1);
    k_gemv_wmma<<<128, 256, 0, stream>>>(w2, hbuf, x2, x3, EMB, 4 * EMB, 0);

    // 9) LM head: 50257 rows (tail tile handled by clamp + predicated store)
    {
        const int tiles = (VOCAB + 15) / 16;             // 3142
        k_gemv_wmma<<<tiles, 256, 0, stream>>>(lm_w, x3, nullptr, logits, VOCAB, EMB, 0);
    }
}